// NESDE_79439715107005
// MI455X (gfx1250) — compile-verified
//
#include <hip/hip_runtime.h>
#include <math.h>

typedef __attribute__((ext_vector_type(2))) float v2f;
typedef __attribute__((ext_vector_type(4))) float v4f;
typedef __attribute__((ext_vector_type(8))) float v8f;

#define WPB 8               // waves (batches) per block
#define MS 20               // padded LDS row stride in floats -> conflict-free banking
#define MATF (16 * MS)      // 320 floats per 16x16 tile
#define LDSW (5 * MATF + 32)  // 1632 floats per wave (5 tiles + scalar pad)

// uniform-lane broadcast via v_readlane (keeps inversion off the DS pipe)
__device__ __forceinline__ float bcast(float v, int l) {
  return __int_as_float(__builtin_amdgcn_readlane(__float_as_int(v), l));
}

// One 16x16x16 fp32 GEMM per wave: C = A * B (or A * B^T), via 4 chained
// V_WMMA_F32_16X16X4_F32. Tiles are row-major with padded stride MS in LDS.
//   A (16x4 f32): M = lane&15; VGPR0/1 hold K = 2*(lane>>4) + {0,1}
//   B (4x16 f32): N = lane&15; VGPR0/1 hold K = 2*(lane>>4) + {0,1}
//   C/D (16x16) : N = lane&15; VGPR v holds M = v + 8*(lane>>4)
__device__ __forceinline__ void gemm16(const float* __restrict__ A,
                                       const float* __restrict__ B,
                                       float* __restrict__ C,
                                       int lane, bool transB) {
  const int r15  = lane & 15;
  const int half = lane >> 4;
  v8f acc = {0.f, 0.f, 0.f, 0.f, 0.f, 0.f, 0.f, 0.f};
#pragma unroll
  for (int k0 = 0; k0 < 16; k0 += 4) {
    const int ka = k0 + 2 * half;
    v2f a, b;
    a.x = A[r15 * MS + ka];
    a.y = A[r15 * MS + ka + 1];
    if (transB) {
      b.x = B[r15 * MS + ka];
      b.y = B[r15 * MS + ka + 1];
    } else {
      b.x = B[ka * MS + r15];
      b.y = B[(ka + 1) * MS + r15];
    }
    acc = __builtin_amdgcn_wmma_f32_16x16x4_f32(false, a, false, b, (short)0,
                                                acc, false, false);
  }
#pragma unroll
  for (int v = 0; v < 8; ++v)
    C[(v + 8 * half) * MS + r15] = acc[v];
}

// integ_exp_trig with t0=0; e must equal exp(lam*t)
__device__ __forceinline__ void integ(float lam, float th, float e, float t,
                                      float& rc, float& rs) {
  float denom = lam * lam + th * th;
  float scale = (denom > 0.f) ? (1.f / fmaxf(denom, 1e-10f)) : 0.f;
  float st, ct;
  __sincosf(th * t, &st, &ct);
  rc = scale * (e * (lam * ct + th * st) - lam);
  rs = scale * (e * (lam * st - th * ct) + th);
}

__global__ __launch_bounds__(32 * WPB) void esde_kernel(
    const float* __restrict__ X0, const float* __restrict__ X0_var,
    const float* __restrict__ U, const float* __restrict__ times,
    const float* __restrict__ Q, const float* __restrict__ lam_r,
    const float* __restrict__ lam_i, const float* __restrict__ Vr,
    const float* __restrict__ Vi, float* __restrict__ out_mean,
    float* __restrict__ out_var, int Bn) {
  __shared__ __align__(16) float lds[WPB * LDSW];
  const int lane = threadIdx.x & 31;
  const int wid  = threadIdx.x >> 5;
  const int b    = blockIdx.x * WPB + wid;
  if (b >= Bn) return;  // whole-wave uniform

  float* PINV = &lds[wid * LDSW];
  float* PHI  = PINV + MATF;
  float* M0   = PINV + 2 * MATF;
  float* M1   = PINV + 3 * MATF;
  float* M2   = PINV + 4 * MATF;
  float* SC   = PINV + 5 * MATF;  // [0:8) lr, [8:16) li, [16:24) e*cos, [24:32) e*sin

  const float t = times[b];

  // ---- issue all bulk global loads up front (latency hides under inversion)
  const v4f* xv4 = (const v4f*)(X0_var + (size_t)b * 256);
  const v4f* qv4 = (const v4f*)(Q + (size_t)b * 256);
  const v4f* vr4 = (const v4f*)(Vr + (size_t)b * 128);
  const v4f* vi4 = (const v4f*)(Vi + (size_t)b * 128);
  v4f xa = xv4[lane], xb = xv4[lane + 32];
  v4f qa = qv4[lane], qb = qv4[lane + 32];
  v4f vra = vr4[lane], via = vi4[lane];

  // ---- per-mode scalars (lanes 0-7) ---------------------------------------
  if (lane < 8) {
    float lr = lam_r[b * 8 + lane];
    float li = lam_i[b * 8 + lane];
    float e  = __expf(t * lr);
    float sv, cv;
    __sincosf(t * li, &sv, &cv);
    SC[lane]      = lr;
    SC[8 + lane]  = li;
    SC[16 + lane] = e * cv;
    SC[24 + lane] = e * sv;
  }

  // ---- single pass over V: build Phi0 (->M0) and Phi (->PHI) --------------
  {
    const int e0 = lane * 4;         // 4 consecutive (i,jj) cells of V
    const int i  = e0 >> 3, jj = e0 & 7;   // jj in {0,4}
    *(v4f*)&M0[i * MS + jj]     = vra;     // Phi0 real block
    *(v4f*)&M0[i * MS + 8 + jj] = via;     // Phi0 imag block
    v4f pr, pi;
#pragma unroll
    for (int c = 0; c < 4; ++c) {
      float A  = SC[16 + jj + c], Bv = SC[24 + jj + c];
      pr[c] = vra[c] * A - via[c] * Bv;
      pi[c] = vra[c] * Bv + via[c] * A;
    }
    *(v4f*)&PHI[i * MS + jj]     = pr;
    *(v4f*)&PHI[i * MS + 8 + jj] = pi;
  }

  // ---- invert Phi0 (register Gauss-Jordan, lane-per-column) -> PINV --------
  float c[16];
#pragma unroll
  for (int i = 0; i < 16; ++i)
    c[i] = (lane < 16) ? M0[i * MS + lane] : ((i == lane - 16) ? 1.f : 0.f);
#pragma unroll
  for (int k = 0; k < 16; ++k) {
    float piv = bcast(c[k], k);
    float ip  = 1.f / piv;
    c[k] *= ip;
#pragma unroll
    for (int i = 0; i < 16; ++i) {
      float f = bcast(c[i], k);      // column-k entry of row i (wave-uniform)
      if (i != k) c[i] -= f * c[k];
    }
  }
  if (lane >= 16) {
#pragma unroll
    for (int i = 0; i < 16; ++i) PINV[i * MS + (lane - 16)] = c[i];
  }

  // ---- variance chain: 8 WMMA GEMMs ---------------------------------------
  {  // stage X0_var -> M1
    int e = lane * 4, i = e >> 4, j = e & 15;
    *(v4f*)&M1[i * MS + j] = xa;
    e += 128; i = e >> 4; j = e & 15;
    *(v4f*)&M1[i * MS + j] = xb;
  }
  gemm16(PINV, M1, M0, lane, false);  // Pinv @ X0_var
  gemm16(M0, PINV, M2, lane, true);   // init_var
  gemm16(PHI, M2, M0, lane, false);   // Phi @ init_var
  gemm16(M0, PHI, M1, lane, true);    // Hom_var -> M1

  {  // stage Q -> M0
    int e = lane * 4, i = e >> 4, j = e & 15;
    *(v4f*)&M0[i * MS + j] = qa;
    e += 128; i = e >> 4; j = e & 15;
    *(v4f*)&M0[i * MS + j] = qb;
  }
  gemm16(PINV, M0, M2, lane, false);  // Pinv @ Q
  gemm16(M2, PINV, M0, lane, true);   // ext_s -> M0

  // ---- sig_int from ext_s (2 pair-cells per lane) -> M2 --------------------
  for (int pp = lane; pp < 64; pp += 32) {
    int i = pp >> 3, j = pp & 7;
    float lamp = -(SC[i] + SC[j]);
    float th1  = SC[8 + i] + SC[8 + j];
    float th2  = SC[8 + j] - SC[8 + i];
    float e    = __expf(lamp * t);
    float c1, s1, c2, s2;
    integ(lamp, th1, e, t, c1, s1);
    integ(lamp, th2, e, t, c2, s2);
    float vcc = 0.5f * (c1 + c2), vsc = 0.5f * (s1 + s2);
    float vcs = 0.5f * (s1 - s2), vss = 0.5f * (c2 - c1);
    float d11 = M0[(2 * i) * MS + 2 * j];
    float d12 = M0[(2 * i) * MS + 2 * j + 1];
    float d21 = M0[(2 * i + 1) * MS + 2 * j];
    float d22 = M0[(2 * i + 1) * MS + 2 * j + 1];
    M2[(2 * i) * MS + 2 * j]         = vcc * d11 - vsc * d21 - vcs * d12 + vss * d22;
    M2[(2 * i) * MS + 2 * j + 1]     = vcs * d11 - vss * d21 + vcc * d12 - vsc * d22;
    M2[(2 * i + 1) * MS + 2 * j]     = vsc * d11 + vcc * d21 - vss * d12 - vcs * d22;
    M2[(2 * i + 1) * MS + 2 * j + 1] = vss * d11 + vcs * d21 + vsc * d12 + vcc * d22;
  }
  gemm16(PHI, M2, M0, lane, false);   // Phi @ sig_int
  gemm16(M0, PHI, M2, lane, true);    // Part_var -> M2

  // ---- write variance: (|diag| fix on Hom_var) + Part_var, b128 stores -----
  {
    v4f* ov4 = (v4f*)(out_var + (size_t)b * 256);
    int e = lane * 4;
#pragma unroll
    for (int ch = 0; ch < 2; ++ch) {
      int i = e >> 4, j = e & 15;
      v4f h = *(v4f*)&M1[i * MS + j];
      v4f p = *(v4f*)&M2[i * MS + j];
      v4f o;
#pragma unroll
      for (int cc = 0; cc < 4; ++cc) {
        float hv = h[cc];
        if (i == j + cc) hv = fabsf(hv);
        o[cc] = hv + p[cc];
      }
      ov4[lane + ch * 32] = o;
      e += 128;
    }
  }

  // ---- mean path: Hom + Part = Phi @ (Pinv@X0 + lambda_int) ----------------
  float x  = (lane < 16) ? X0[b * 16 + lane] : 0.f;
  float uu = (lane < 16) ? U[b * 16 + lane] : 0.f;
  float w = 0.f, u2 = 0.f;
#pragma unroll
  for (int j = 0; j < 16; ++j) {
    float xj = bcast(x, j);
    float uj = bcast(uu, j);
    float pv = (lane < 16) ? PINV[(lane & 15) * MS + j] : 0.f;
    w  += pv * xj;
    u2 += pv * uj;
  }
  int   j2 = (lane & 15) >> 1;
  float lr = SC[j2], li = SC[8 + j2];
  float lamp = -lr;
  float e  = __expf(lamp * t);
  float ci, si;
  integ(lamp, li, e, t, ci, si);
  float up = __shfl_xor(u2, 1, 32);
  float z  = ((lane & 1) == 0) ? (w + ci * u2 - si * up)
                               : (w + si * up + ci * u2);
  float res = 0.f;
#pragma unroll
  for (int j = 0; j < 16; ++j) {
    float zj = bcast(z, j);
    float pv = (lane < 16) ? PHI[(lane & 15) * MS + j] : 0.f;
    res += pv * zj;
  }
  if (lane < 16) out_mean[b * 16 + lane] = res;
}

extern "C" void kernel_launch(void* const* d_in, const int* in_sizes, int n_in,
                              void* d_out, int out_size, void* d_ws, size_t ws_size,
                              hipStream_t stream) {
  const float* X0     = (const float*)d_in[0];
  const float* X0_var = (const float*)d_in[1];
  const float* U      = (const float*)d_in[2];
  const float* times  = (const float*)d_in[3];
  const float* Q      = (const float*)d_in[4];
  const float* lam_r  = (const float*)d_in[5];
  const float* lam_i  = (const float*)d_in[6];
  const float* Vr     = (const float*)d_in[7];
  const float* Vi     = (const float*)d_in[8];
  const int Bn = in_sizes[3];                    // times is (B,)
  float* out_mean = (float*)d_out;               // (B,16,1) flat
  float* out_var  = out_mean + (size_t)Bn * 16;  // (B,16,16) flat
  const int blocks = (Bn + WPB - 1) / WPB;
  esde_kernel<<<blocks, 32 * WPB, 0, stream>>>(
      X0, X0_var, U, times, Q, lam_r, lam_i, Vr, Vi, out_mean, out_var, Bn);
}